// LMulLinear_62234076119648
// MI455X (gfx1250) — compile-verified
//
#include <hip/hip_runtime.h>

typedef __attribute__((ext_vector_type(2))) float v2f;
typedef __attribute__((ext_vector_type(8))) float v8f;

#define M_DIM 256
#define N_DIM 512
#define K_DIM 512
#define BM 64
#define BN 64
#define BK 32
#define LDW 36                 // padded row stride in floats: 144B rows (16B aligned),
                               // 36*i mod 64 = 4*(9i mod 16) -> conflict-free b64 frag reads
#define TILE_ELEMS (BM * LDW)  // one 64x32 tile with padding (9216 B)
#define BUF_ELEMS (2 * TILE_ELEMS)  // A tile + W tile per pipeline buffer

// u(v) = round(mant*2^4) * 2^(2*ex - 4), v = mant*2^ex, mant in [0.5,1).
// Branch-free bit formulation: round the mantissa to 3 bits in the raw encoding
// with ties-to-even (qv = RNE-quantized v; exponent carry folds in naturally),
// then u = qv * 2^(e-126) using the ORIGINAL biased exponent e.
//   no carry: qv = qm*2^(ex-1), u = qm*2^(2ex-1) = round(m*16)*2^(2ex-4)  (qm in [1,2))
//   carry   : qv = 2^ex,        u = 2^(2ex)     = 8*2^(2ex-3)            ✓
//   v == 0  : qv = 0 -> u = 0                                            ✓
__device__ __forceinline__ float lmul_u(float v) {
  const unsigned int b = __float_as_uint(v);
  const unsigned int r = b + 0x0007FFFFu + ((b >> 20) & 1u);  // RNE at mantissa bit 20
  const float qv = __uint_as_float(r & 0xFFF00000u);          // sign|exp|3 mantissa bits
  const int e = (int)((b >> 23) & 0xFFu);
  return ldexpf(qv, e - 126);
}

// Async-DMA one 64x32 raw tile of x and of w into the LDS buffer at byte offset
// `abase` (A region) / `abase + TILE_ELEMS*4` (W region). b128 per lane, 16B aligned.
__device__ __forceinline__ void issue_async_tile(const float* __restrict__ x,
                                                 const float* __restrict__ w,
                                                 int gm0, int gn0, int kt,
                                                 unsigned abase, int tid) {
#pragma unroll
  for (int i = 0; i < 4; ++i) {
    const int idx = tid + i * 128;  // 0..511 float4 chunks over a 64x32 tile
    const int row = idx >> 3;       // 0..63
    const int c4 = (idx & 7) << 2;  // 0,4,...,28
    const unsigned aoff = abase + (unsigned)((row * LDW + c4) * 4);
    const unsigned long long ga =
        (unsigned long long)(uintptr_t)(x + (gm0 + row) * K_DIM + kt + c4);
    asm volatile("global_load_async_to_lds_b128 %0, %1, off"
                 :: "v"(aoff), "v"(ga) : "memory");
    const unsigned woff = aoff + (unsigned)(TILE_ELEMS * 4);
    const unsigned long long gw =
        (unsigned long long)(uintptr_t)(w + (gn0 + row) * K_DIM + kt + c4);
    asm volatile("global_load_async_to_lds_b128 %0, %1, off"
                 :: "v"(woff), "v"(gw) : "memory");
  }
}

// In-place u-transform of both tiles; each thread owns its own float4 slots,
// so there is no intra-pass hazard. Keeps all VALU off the WMMA critical path.
__device__ __forceinline__ void transform_tile(float* buf, int tid) {
#pragma unroll
  for (int i = 0; i < 4; ++i) {
    const int idx = tid + i * 128;
    const int row = idx >> 3;
    const int c4 = (idx & 7) << 2;
    float4* pa = (float4*)&buf[row * LDW + c4];
    float4 va = *pa;
    va.x = lmul_u(va.x);
    va.y = lmul_u(va.y);
    va.z = lmul_u(va.z);
    va.w = lmul_u(va.w);
    *pa = va;
    float4* pw = (float4*)&buf[TILE_ELEMS + row * LDW + c4];
    float4 vw = *pw;
    vw.x = lmul_u(vw.x);
    vw.y = lmul_u(vw.y);
    vw.z = lmul_u(vw.z);
    vw.w = lmul_u(vw.w);
    *pw = vw;
  }
}

// Pure ds_load -> v_wmma_f32_16x16x4_f32 K-loop over one buffered tile pair.
__device__ __forceinline__ void compute_tile(const float* buf, v8f acc[4],
                                             int mrow, int koff, int cbn) {
#pragma unroll
  for (int k0 = 0; k0 < BK; k0 += 4) {
    const int kb = k0 + koff;
    const v2f a = *(const v2f*)&buf[mrow * LDW + kb];
#pragma unroll
    for (int nt = 0; nt < 4; ++nt) {
      const v2f b = *(const v2f*)&buf[TILE_ELEMS + (nt * 16 + cbn) * LDW + kb];
      acc[nt] = __builtin_amdgcn_wmma_f32_16x16x4_f32(
          /*neg_a=*/false, a, /*neg_b=*/false, b,
          /*c_mod=*/(short)0, acc[nt], /*reuse_a=*/false, /*reuse_b=*/false);
    }
  }
}

__global__ __launch_bounds__(128) void lmul_linear_wmma_pipe(
    const float* __restrict__ x,     // (256, 512)
    const float* __restrict__ w,     // (512, 512) row-major (OUT, IN)
    const float* __restrict__ bias,  // (512,)
    float* __restrict__ out) {       // (256, 512)
  __shared__ float smem[2 * BUF_ELEMS];  // double-buffered {A tile, W tile}

  const int tid = threadIdx.x;
  const int lane = tid & 31;
  const int wv = tid >> 5;  // wave 0..3
  const int gm0 = blockIdx.y * BM;
  const int gn0 = blockIdx.x * BN;

  // byte offset of smem within the workgroup's LDS allocation (low 32 bits of the
  // generic pointer are the LDS offset on AMDGPU)
  const unsigned lds_base = (unsigned)(unsigned long long)(uintptr_t)smem;

  // A fragment (32-bit 16x4): lanes 0-15 -> K = k0,k0+1 ; lanes 16-31 -> K = k0+2,k0+3
  const int mrow = wv * 16 + (lane & 15);
  const int koff = (lane >> 4) << 1;  // 0 or 2
  const int cbn = lane & 15;          // column within a 16-wide N subtile

  v8f acc[4] = {v8f{}, v8f{}, v8f{}, v8f{}};

  // ---- software pipeline: DMA(t+1) overlaps transform+WMMA of tile t ----
  issue_async_tile(x, w, gm0, gn0, 0, lds_base, tid);
  const int T = K_DIM / BK;  // 16 k-tiles
#pragma unroll 2             // makes buffer parity compile-time constant
  for (int t = 0; t < T; ++t) {
    const int cur = t & 1;
    float* buf = &smem[cur * BUF_ELEMS];
    const unsigned nextbase = lds_base + (unsigned)(((cur ^ 1) * BUF_ELEMS) * 4);

    asm volatile("s_wait_asynccnt 0" ::: "memory");  // my DMA into buf done
    __syncthreads();  // everyone's DMA done; all reads of the other buffer done
    if (t + 1 < T) issue_async_tile(x, w, gm0, gn0, (t + 1) * BK, nextbase, tid);
    transform_tile(buf, tid);
    __syncthreads();  // transformed tile visible to all waves
    compute_tile(buf, acc, mrow, koff, cbn);
  }

  // ---- epilogue: C/D layout (VGPR r -> M = r + 8*(lane>=16), N = lane&15) ----
  const int r0 = (lane >> 4) * 8;
#pragma unroll
  for (int nt = 0; nt < 4; ++nt) {
    const int gcol = gn0 + nt * 16 + cbn;
    const float bv = bias[gcol];
#pragma unroll
    for (int r = 0; r < 8; ++r) {
      const int grow = gm0 + wv * 16 + r0 + r;
      out[grow * N_DIM + gcol] = acc[nt][r] + bv;
    }
  }
}

extern "C" void kernel_launch(void* const* d_in, const int* in_sizes, int n_in,
                              void* d_out, int out_size, void* d_ws, size_t ws_size,
                              hipStream_t stream) {
  (void)in_sizes; (void)n_in; (void)d_ws; (void)ws_size; (void)out_size;
  const float* x = (const float*)d_in[0];     // (2,128,512) -> (256,512)
  const float* w = (const float*)d_in[1];     // (512,512)
  const float* bias = (const float*)d_in[2];  // (512,)
  float* out = (float*)d_out;                 // (256,512)
  dim3 grid(N_DIM / BN, M_DIM / BM);          // (8, 4) = 32 blocks
  lmul_linear_wmma_pipe<<<grid, dim3(128, 1, 1), 0, stream>>>(x, w, bias, out);
}